// PureRNN_53815940219132
// MI455X (gfx1250) — compile-verified
//
#include <hip/hip_runtime.h>
#include <math.h>

// Problem constants
#define B_  64
#define T_  512
#define D_  1024
#define H_  1024
#define G_  4096      // 4*H
#define L_  4
#define TC  128       // timestep chunk
#define NCHUNK (T_/TC)
#define KC  64        // K-tile for input GEMM

typedef float v2f __attribute__((ext_vector_type(2)));
typedef float v8f __attribute__((ext_vector_type(8)));

__device__ __forceinline__ v8f wmma_f32(v2f a, v2f b, v8f c) {
  // D = A(16x4, f32) * B(4x16, f32) + C(16x16, f32)
  return __builtin_amdgcn_wmma_f32_16x16x4_f32(
      /*neg_a=*/false, a, /*neg_b=*/false, b,
      /*c_mod=*/(short)0, c, /*reuse_a=*/false, /*reuse_b=*/false);
}

// ---------------------------------------------------------------------------
// Kernel 1: input-side GEMM for one layer / one T-chunk.
//   xg[(tloc*64 + b), g] = sum_k in[t,b,k] * Wih[g,k] + bih[g] + bhh[g]
// Grid: (G/64, TC). Block: 256 threads (8 waves). 64x64 output tile per WG.
// LDS-tiled fp32 WMMA; each wave computes two 16x16 tiles sharing A.
// global_prefetch of the next K-tile hides HBM latency behind the WMMA loop.
// ---------------------------------------------------------------------------
__global__ __launch_bounds__(256) void lstm_xgemm(
    const float* __restrict__ in,   // layer input (x: [B,T,D] if layer0 else [T,B,H])
    const float* __restrict__ Wih,  // [4096,1024] this layer
    const float* __restrict__ bih,  // [4096]
    const float* __restrict__ bhh,  // [4096]
    float* __restrict__ xg,         // [TC*64, 4096]
    int chunk, int layer0)
{
  __shared__ __align__(16) float Al[64][KC + 2];
  __shared__ __align__(16) float Bl[64][KC + 2];

  const int tid  = threadIdx.x;
  const int lane = tid & 31;
  const int wv   = tid >> 5;     // wave 0..7
  const int mt   = wv & 3;       // M subtile (16 rows)
  const int np   = wv >> 2;      // N pair: cols np*32 .. np*32+31
  const int ln16 = lane & 15;
  const int lhi  = lane >> 4;

  const int tloc  = blockIdx.y;           // 0..TC-1
  const int tglob = chunk * TC + tloc;    // global timestep
  const int col0  = blockIdx.x * 64;      // gate-space column base

  // bias replicated down the M dimension of the accumulator tile
  const float bias0 = bih[col0 + np*32 +      ln16] + bhh[col0 + np*32 +      ln16];
  const float bias1 = bih[col0 + np*32 + 16 + ln16] + bhh[col0 + np*32 + 16 + ln16];
  v8f acc0, acc1;
#pragma unroll
  for (int r = 0; r < 8; ++r) { acc0[r] = bias0; acc1[r] = bias1; }

  for (int k0 = 0; k0 < D_; k0 += KC) {
    // Stage A (activations) tile: rows b=0..63, cols k0..k0+KC
    for (int i = tid; i < 64 * (KC/4); i += 256) {
      const int b  = i >> 4;
      const int c4 = (i & 15) << 2;
      const float* src = layer0
          ? (in + ((size_t)b * T_ + tglob) * D_ + k0 + c4)     // x[b, t, k]
          : (in + ((size_t)tglob * B_ + b) * H_ + k0 + c4);    // seq[t, b, k]
      const float4 v = *(const float4*)src;
      if (k0 + KC < D_) __builtin_prefetch(src + KC, 0, 1);    // next K-tile
      Al[b][c4+0] = v.x; Al[b][c4+1] = v.y; Al[b][c4+2] = v.z; Al[b][c4+3] = v.w;
    }
    // Stage B (weights) tile: Wih rows col0..col0+63
    for (int i = tid; i < 64 * (KC/4); i += 256) {
      const int j  = i >> 4;
      const int c4 = (i & 15) << 2;
      const float* src = Wih + (size_t)(col0 + j) * D_ + k0 + c4;
      const float4 v = *(const float4*)src;
      if (k0 + KC < D_) __builtin_prefetch(src + KC, 0, 1);    // next K-tile
      Bl[j][c4+0] = v.x; Bl[j][c4+1] = v.y; Bl[j][c4+2] = v.z; Bl[j][c4+3] = v.w;
    }
    __syncthreads();
#pragma unroll
    for (int kk = 0; kk < KC; kk += 4) {
      const int ko = kk + 2*lhi;
      v2f a  = *(const v2f*)&Al[mt*16 + ln16][ko];
      v2f b0 = *(const v2f*)&Bl[np*32 +      ln16][ko];
      v2f b1 = *(const v2f*)&Bl[np*32 + 16 + ln16][ko];
      acc0 = wmma_f32(a, b0, acc0);
      acc1 = wmma_f32(a, b1, acc1);
    }
    __syncthreads();
  }

  // Store: C layout -> M = r + 8*(lane>=16), N = lane%16
#pragma unroll
  for (int r = 0; r < 8; ++r) {
    const int b = mt*16 + r + lhi*8;
    const size_t row = (size_t)tloc * 64 + b;
    xg[row * G_ + col0 + np*32 +      ln16] = acc0[r];
    xg[row * G_ + col0 + np*32 + 16 + ln16] = acc1[r];
  }
}

// ---------------------------------------------------------------------------
// Kernel 2: persistent recurrent scan over one T-chunk.
// 64 WGs; WG w owns h-columns [16w, 16w+16). Whh rows for all 4 gates of
// those columns (64 rows x 1024 f32 = 256KB) live in LDS for the whole chunk.
// Per step: z = xg_t + h @ Whh^T via fp32 WMMA, gates in LDS, c in LDS,
// h ping-ponged through global, software grid barrier between steps.
// Next step's xg tile is prefetched behind the current GEMM.
// ---------------------------------------------------------------------------
__global__ __launch_bounds__(256) void lstm_recurrent(
    const float* __restrict__ xg,     // [TC*64, 4096]
    const float* __restrict__ Whh,    // [4096,1024] this layer
    float* __restrict__ h0,           // [64,1024] ping
    float* __restrict__ h1,           // [64,1024] pong
    float* __restrict__ cbuf,         // [64,1024]
    float* __restrict__ seqOut,       // [T,64,1024]
    unsigned* __restrict__ cnt,       // grid barrier counter (zeroed by host)
    int chunk)
{
  extern __shared__ float lds[];
  float* Wl = lds;                    // [64][1026]  (padded rows)
  float* zl = Wl + 64 * 1026;         // [4][64][16]
  float* cl = zl + 4 * 64 * 16;       // [64][16]

  const int tid  = threadIdx.x;
  const int lane = tid & 31;
  const int wv   = tid >> 5;
  const int mt   = wv & 3;            // batch subtile: rows 16*mt..+16
  const int gp   = wv >> 2;           // gate pair: 0 -> {i,f}, 1 -> {g,o}
  const int ln16 = lane & 15;
  const int lhi  = lane >> 4;
  const int j0   = blockIdx.x * 16;   // h-column base

  // Load Whh rows for our columns into LDS (row r = gate*16 + jj)
  for (int i = tid; i < 64 * (H_/4); i += 256) {
    const int r  = i >> 8;
    const int c4 = (i & 255) << 2;
    const int gate = r >> 4, jj = r & 15;
    const float4 v = *(const float4*)(Whh + ((size_t)gate * H_ + j0 + jj) * H_ + c4);
    float* dst = Wl + r * 1026 + c4;
    dst[0] = v.x; dst[1] = v.y; dst[2] = v.z; dst[3] = v.w;
  }
  // Load carried cell state
  for (int i = tid; i < 64 * 16; i += 256)
    cl[i] = cbuf[(size_t)(i >> 4) * H_ + j0 + (i & 15)];
  __syncthreads();

  const float* b0ptr = Wl + ((2*gp    ) * 16 + ln16) * 1026;
  const float* b1ptr = Wl + ((2*gp + 1) * 16 + ln16) * 1026;

  for (int s = 0; s < TC; ++s) {
    const int tt = chunk * TC + s;
    const float* hin  = (tt & 1) ? h1 : h0;
    float*       hout = (tt & 1) ? h0 : h1;

    // acc init = xg tile (bias already folded in by lstm_xgemm)
    v8f acc0, acc1;
    {
      const size_t base = (size_t)s * 64 * G_;
#pragma unroll
      for (int r = 0; r < 8; ++r) {
        const int b = mt*16 + r + lhi*8;
        acc0[r] = xg[base + (size_t)b * G_ + (2*gp    ) * H_ + j0 + ln16];
        acc1[r] = xg[base + (size_t)b * G_ + (2*gp + 1) * H_ + j0 + ln16];
      }
    }
    // Prefetch next step's xg tile behind this step's GEMM
    if (s + 1 < TC) {
      const size_t nbase = (size_t)(s + 1) * 64 * G_;
#pragma unroll
      for (int r = 0; r < 8; ++r) {
        const int b = mt*16 + r + lhi*8;
        __builtin_prefetch(xg + nbase + (size_t)b * G_ + (2*gp    ) * H_ + j0 + ln16, 0, 1);
        __builtin_prefetch(xg + nbase + (size_t)b * G_ + (2*gp + 1) * H_ + j0 + ln16, 0, 1);
      }
    }

    // z += h @ Whh^T  (A from global h, B from LDS weights)
    const float* aptr = hin + (size_t)(mt*16 + ln16) * H_;
#pragma unroll 4
    for (int k = 0; k < H_; k += 4) {
      const int kk = k + 2*lhi;
      v2f a  = *(const v2f*)(aptr  + kk);
      v2f b0 = *(const v2f*)(b0ptr + kk);
      v2f b1 = *(const v2f*)(b1ptr + kk);
      acc0 = wmma_f32(a, b0, acc0);
      acc1 = wmma_f32(a, b1, acc1);
    }

    // Spill gate tiles to LDS for the cross-wave elementwise phase
#pragma unroll
    for (int r = 0; r < 8; ++r) {
      const int b = mt*16 + r + lhi*8;
      zl[((2*gp    ) * 64 + b) * 16 + ln16] = acc0[r];
      zl[((2*gp + 1) * 64 + b) * 16 + ln16] = acc1[r];
    }
    __syncthreads();

    // LSTM cell update: i,f,g,o -> c,h  (torch gate order)
#pragma unroll
    for (int e = 0; e < 4; ++e) {
      const int idx = tid * 4 + e;           // b*16 + n
      const int b = idx >> 4, n = idx & 15;
      const float zi = zl[(0*64 + b) * 16 + n];
      const float zf = zl[(1*64 + b) * 16 + n];
      const float zg = zl[(2*64 + b) * 16 + n];
      const float zo = zl[(3*64 + b) * 16 + n];
      const float ig = 1.0f / (1.0f + __expf(-zi));
      const float fg = 1.0f / (1.0f + __expf(-zf));
      const float gg = tanhf(zg);
      const float og = 1.0f / (1.0f + __expf(-zo));
      const float c  = fg * cl[idx] + ig * gg;
      cl[idx] = c;
      const float h = og * tanhf(c);
      hout[(size_t)b * H_ + j0 + n] = h;
      seqOut[((size_t)tt * B_ + b) * H_ + j0 + n] = h;
    }

    // ---- software grid barrier (release h, wait for all 64 WGs) ----
    __threadfence();          // every thread: flush h stores to device scope
    __syncthreads();
    if (tid == 0) {
      atomicAdd(cnt, 1u);
      const unsigned target = (unsigned)gridDim.x * (unsigned)(s + 1);
      while (__hip_atomic_load(cnt, __ATOMIC_ACQUIRE, __HIP_MEMORY_SCOPE_AGENT) < target) {
        __builtin_amdgcn_s_sleep(2);
      }
    }
    __syncthreads();
    __threadfence();          // every thread: acquire before reading new h
  }

  // Persist cell state across chunk launches
  for (int i = tid; i < 64 * 16; i += 256)
    cbuf[(size_t)(i >> 4) * H_ + j0 + (i & 15)] = cl[i];
}

// ---------------------------------------------------------------------------
// Host-side orchestration (graph-capture safe: only async ops on `stream`).
// ---------------------------------------------------------------------------
extern "C" void kernel_launch(void* const* d_in, const int* in_sizes, int n_in,
                              void* d_out, int out_size, void* d_ws, size_t ws_size,
                              hipStream_t stream)
{
  const float* x   = (const float*)d_in[0];  // [B,T,D]
  const float* Wih = (const float*)d_in[1];  // [L,4H,D]
  const float* Whh = (const float*)d_in[2];  // [L,4H,H]
  const float* bih = (const float*)d_in[3];  // [L,4H]
  const float* bhh = (const float*)d_in[4];  // [L,4H]

  char* ws = (char*)d_ws;
  const size_t SEQ_BYTES = (size_t)T_ * B_ * H_ * sizeof(float);   // 128 MB
  const size_t XG_BYTES  = (size_t)TC * B_ * G_ * sizeof(float);   // 128 MB
  const size_t H_BYTES   = (size_t)B_ * H_ * sizeof(float);        // 256 KB

  float*    seqA = (float*)(ws);
  float*    seqB = (float*)(ws + SEQ_BYTES);
  float*    xg   = (float*)(ws + 2 * SEQ_BYTES);
  float*    h0   = (float*)(ws + 2 * SEQ_BYTES + XG_BYTES);
  float*    h1   = (float*)(ws + 2 * SEQ_BYTES + XG_BYTES + H_BYTES);
  float*    cb   = (float*)(ws + 2 * SEQ_BYTES + XG_BYTES + 2 * H_BYTES);
  unsigned* cnt  = (unsigned*)(ws + 2 * SEQ_BYTES + XG_BYTES + 3 * H_BYTES);

  const size_t dynLds = (size_t)(64 * 1026 + 4 * 64 * 16 + 64 * 16) * sizeof(float); // ~277 KB

  const float* layerIn  = x;
  float*       layerOut = seqA;
  for (int l = 0; l < L_; ++l) {
    hipMemsetAsync(h0, 0, H_BYTES, stream);
    hipMemsetAsync(h1, 0, H_BYTES, stream);
    hipMemsetAsync(cb, 0, H_BYTES, stream);
    for (int chunk = 0; chunk < NCHUNK; ++chunk) {
      lstm_xgemm<<<dim3(G_/64, TC), 256, 0, stream>>>(
          layerIn,
          Wih + (size_t)l * G_ * D_,
          bih + (size_t)l * G_,
          bhh + (size_t)l * G_,
          xg, chunk, (l == 0) ? 1 : 0);
      hipMemsetAsync(cnt, 0, sizeof(unsigned), stream);
      lstm_recurrent<<<dim3(H_/16), 256, dynLds, stream>>>(
          xg,
          Whh + (size_t)l * G_ * H_,
          h0, h1, cb, layerOut, cnt, chunk);
    }
    layerIn  = layerOut;
    layerOut = (layerOut == seqA) ? seqB : seqA;
  }
  // Final h of last layer: step tt=511 wrote hbuf[(511+1)&1] = h0
  hipMemcpyAsync(d_out, h0, H_BYTES, hipMemcpyDeviceToDevice, stream);
}